// SAGE_42992622633736
// MI455X (gfx1250) — compile-verified
//
#include <hip/hip_runtime.h>

typedef __attribute__((ext_vector_type(2))) float v2f;
typedef __attribute__((ext_vector_type(8))) float v8f;

#define N_NODES 100000
#define E_EDGES 1000000
#define NTILES  (N_NODES / 16)   // 6250, exact
#define WPB     8                // waves per block (256 threads)

// ---------------- zero workspace ----------------
__global__ void sage_zero(float* __restrict__ p, int n4) {
    int i = blockIdx.x * blockDim.x + threadIdx.x;
    int stride = gridDim.x * blockDim.x;
    float4 z = make_float4(0.f, 0.f, 0.f, 0.f);
    for (; i < n4; i += stride) ((float4*)p)[i] = z;
}

// ---------------- edge gather + scatter-add (mean aggregation numerator) ----
// 16 threads per edge, each moves a float4 slice of the 64-float row.
__global__ void sage_scatter(const float* __restrict__ x,
                             const int*   __restrict__ src,
                             const int*   __restrict__ dst,
                             float*       __restrict__ agg,
                             float*       __restrict__ deg,
                             int count_deg) {
    int idx = blockIdx.x * blockDim.x + threadIdx.x;
    if (idx >= E_EDGES * 16) return;
    int e = idx >> 4;
    int f = (idx & 15) << 2;
    int s = src[e];
    int d = dst[e];
    float4 v = *(const float4*)(x + (size_t)s * 64 + f);
    float* a = agg + (size_t)d * 64 + f;
    atomicAdd(a + 0, v.x);
    atomicAdd(a + 1, v.y);
    atomicAdd(a + 2, v.z);
    atomicAdd(a + 3, v.w);
    if (count_deg && f == 0) atomicAdd(deg + d, 1.0f);
}

// ---------------- fused SAGE layer: out = act(x@Ws + (agg/deg)@Wn + b) ------
// One wave computes a 16-node tile with V_WMMA_F32_16X16X4_F32.
// A layout (16x4 f32): lanes 0-15 -> M=lane {K=0,1}, lanes 16-31 -> M=lane-16 {K=2,3}.
// C/D layout (16x16 f32): VGPR j, lanes 0-15 -> M=j, lanes 16-31 -> M=j+8; N=lane&15.
template <int NOUT, bool RELU>
__global__ __launch_bounds__(256) void sage_layer(
    const float* __restrict__ x,      // [N,64]
    const float* __restrict__ agg,    // [N,64]
    const float* __restrict__ deg,    // [N]
    const float* __restrict__ Wself,  // [64,NOUT] row-major
    const float* __restrict__ Wneigh, // [64,NOUT] row-major
    const float* __restrict__ bias,   // [NOUT]
    float*       __restrict__ out) {  // [N,NOUT]
    constexpr int NT = NOUT / 16;

    // Stage weights transposed ([n][k]) so B fragments are contiguous 8B LDS loads.
    __shared__ float sWs[NOUT * 64];
    __shared__ float sWn[NOUT * 64];
    __shared__ float sB[NOUT];
    for (int i = threadIdx.x; i < 64 * NOUT; i += blockDim.x) {
        int k = i / NOUT, n = i % NOUT;
        sWs[n * 64 + k] = Wself[i];
        sWn[n * 64 + k] = Wneigh[i];
    }
    for (int i = threadIdx.x; i < NOUT; i += blockDim.x) sB[i] = bias[i];
    __syncthreads();

    int wave = threadIdx.x >> 5;
    int tile = blockIdx.x * WPB + wave;
    if (tile >= NTILES) return;   // wave-uniform: EXEC stays all-ones for WMMA

    int lane = threadIdx.x & 31;
    int half = lane >> 4;   // selects K pair (A/B) and M+8 (C/D)
    int lm   = lane & 15;   // M row (A) / N column (B, C/D)
    int row  = tile * 16 + lm;

    float dv  = deg[row];
    float inv = 1.0f / fmaxf(dv, 1.0f);   // zero-degree -> mean contributes 0

    v8f c[NT];
#pragma unroll
    for (int nt = 0; nt < NT; ++nt) c[nt] = (v8f){};

    const float* xr = x + (size_t)row * 64;
    const float* ar = agg + (size_t)row * 64;

#pragma unroll
    for (int kk = 0; kk < 16; ++kk) {
        int k0 = kk * 4 + half * 2;
        v2f as = *(const v2f*)(xr + k0);
        v2f am = *(const v2f*)(ar + k0);
        am = am * inv;   // fold mean division into the A fragment
#pragma unroll
        for (int nt = 0; nt < NT; ++nt) {
            v2f bs = *(const v2f*)(&sWs[(nt * 16 + lm) * 64 + k0]);
            v2f bn = *(const v2f*)(&sWn[(nt * 16 + lm) * 64 + k0]);
            c[nt] = __builtin_amdgcn_wmma_f32_16x16x4_f32(
                false, as, false, bs, (short)0, c[nt], false, false);
            c[nt] = __builtin_amdgcn_wmma_f32_16x16x4_f32(
                false, am, false, bn, (short)0, c[nt], false, false);
        }
    }

#pragma unroll
    for (int nt = 0; nt < NT; ++nt) {
        int col = nt * 16 + lm;
        float bb = sB[col];
#pragma unroll
        for (int j = 0; j < 8; ++j) {
            int r = tile * 16 + j + half * 8;
            float v = c[nt][j] + bb;
            if (RELU) v = fmaxf(v, 0.0f);
            out[(size_t)r * NOUT + col] = v;
        }
    }
}

extern "C" void kernel_launch(void* const* d_in, const int* in_sizes, int n_in,
                              void* d_out, int out_size, void* d_ws, size_t ws_size,
                              hipStream_t stream) {
    const float* features = (const float*)d_in[0];
    const int*   src      = (const int*)d_in[1];
    const int*   dst      = (const int*)d_in[2];
    const float* w_self1  = (const float*)d_in[3];
    const float* w_neigh1 = (const float*)d_in[4];
    const float* b1       = (const float*)d_in[5];
    const float* w_self2  = (const float*)d_in[6];
    const float* w_neigh2 = (const float*)d_in[7];
    const float* b2       = (const float*)d_in[8];
    float* out = (float*)d_out;

    float* ws   = (float*)d_ws;
    float* agg1 = ws;                              // N*64
    float* agg2 = agg1 + (size_t)N_NODES * 64;     // N*64
    float* deg  = agg2 + (size_t)N_NODES * 64;     // N
    float* h    = deg + N_NODES;                   // N*64

    // 1) zero agg1 | agg2 | deg (contiguous region)
    int zero4 = (N_NODES * 64 * 2 + N_NODES + 3) / 4;
    sage_zero<<<4096, 256, 0, stream>>>(ws, zero4);

    // 2) layer-1 neighbor sum + degree
    int sblocks = (E_EDGES * 16 + 255) / 256;
    sage_scatter<<<sblocks, 256, 0, stream>>>(features, src, dst, agg1, deg, 1);

    // 3) layer 1: h = relu(x@Ws1 + mean@Wn1 + b1)
    int lblocks = (NTILES + WPB - 1) / WPB;
    sage_layer<64, true><<<lblocks, 256, 0, stream>>>(
        features, agg1, deg, w_self1, w_neigh1, b1, h);

    // 4) layer-2 neighbor sum over h
    sage_scatter<<<sblocks, 256, 0, stream>>>(h, src, dst, agg2, deg, 0);

    // 5) layer 2: out = h@Ws2 + mean@Wn2 + b2
    sage_layer<32, false><<<lblocks, 256, 0, stream>>>(
        h, agg2, deg, w_self2, w_neigh2, b2, out);
}